// TransE_score_22617297780848
// MI455X (gfx1250) — compile-verified
//
#include <hip/hip_runtime.h>
#include <stdint.h>

// TransE L1 score: score[b,n] = GAMMA - sum_d |(sub[b,d]+rel[b,d]) - x[n,d]|
// B=48, D=256 fixed by reference; N, K derived from in_sizes.

#define GAMMA_F 12.0f

constexpr int Bdim      = 48;
constexpr int Ddim      = 256;
constexpr int NT        = 32;    // entities per tile (= lanes per wave)
constexpr int XSTRIDE   = 260;   // padded LDS row stride in floats:
                                 // 260 % 64 == 4 -> b128 reads hit all banks (2-way max)
constexpr int TPB_TILES = 2;     // tiles per block, double-buffered async staging

// Issue the async global->LDS stage of one [NT x 256] f32 x-tile (8 b128 per thread).
__device__ __forceinline__ void issue_xtile_async(
    const float* __restrict__ x, int n0, int N, float* xsbuf, int tid)
{
    const uint64_t gbase = (uint64_t)(uintptr_t)x + (uint64_t)n0 * (Ddim * 4);
    #pragma unroll
    for (int k = 0; k < (NT * Ddim / 4) / 256; ++k) {
        const int g   = tid + k * 256;    // float4 index within tile
        const int row = g >> 6;           // / (Ddim/4)
        const int c4  = g & 63;           // float4 column
        if (n0 + row < N) {
            // generic LDS pointer: low 32 bits == LDS byte offset
            const uint32_t lds_addr =
                (uint32_t)(uintptr_t)(&xsbuf[row * XSTRIDE + c4 * 4]);
            const uint32_t goff = (uint32_t)(row * (Ddim * 4) + c4 * 16);
            asm volatile("global_load_async_to_lds_b128 %0, %1, %2"
                         :: "v"(lds_addr), "v"(goff), "s"(gbase)
                         : "memory");
        }
    }
}

__global__ __launch_bounds__(256) void transe_score_kernel(
    const float* __restrict__ sub, const float* __restrict__ rel,
    const float* __restrict__ x, float* __restrict__ score, int N)
{
    __shared__ __align__(16) float objs[Bdim * Ddim];       // 48 KB: obj = sub + rel
    __shared__ __align__(16) float xs[2][NT * XSTRIDE];     // 2 x ~33 KB: x tile ping-pong

    const int tid   = threadIdx.x;
    const int tile0 = blockIdx.x * TPB_TILES;

    // ---- obj = sub + rel into LDS, vectorized (12 float4 iters per thread)
    {
        const float4* __restrict__ s4 = (const float4*)sub;
        const float4* __restrict__ r4 = (const float4*)rel;
        float4* o4 = (float4*)objs;
        #pragma unroll
        for (int k = 0; k < (Bdim * Ddim / 4) / 256; ++k) {
            const int i = tid + k * 256;
            const float4 a = s4[i];
            const float4 r = r4[i];
            o4[i] = make_float4(a.x + r.x, a.y + r.y, a.z + r.z, a.w + r.w);
        }
    }

    // ---- prefetch first x tile via the async engine
    issue_xtile_async(x, tile0 * NT, N, xs[0], tid);

    const int lane = tid & 31;   // entity within tile
    const int wid  = tid >> 5;   // 0..7, b-phase

    for (int ti = 0; ti < TPB_TILES; ++ti) {
        const int n0 = (tile0 + ti) * NT;

        // All of this wave's async loads done, then block-wide: everyone's loads
        // for buffer ti&1 are visible AND everyone finished reading buffer (ti+1)&1
        // (computed last iteration) -> safe to start overwriting it below.
        asm volatile("s_wait_asynccnt 0" ::: "memory");
        __syncthreads();

        if (ti + 1 < TPB_TILES)
            issue_xtile_async(x, (tile0 + ti + 1) * NT, N, xs[(ti + 1) & 1], tid);

        // ---- compute tile ti out of LDS: 6 b-rows per thread, x row cached in regs
        float4 acc[6];
        #pragma unroll
        for (int i = 0; i < 6; ++i) acc[i] = make_float4(0.f, 0.f, 0.f, 0.f);

        const float4* __restrict__ xrow = (const float4*)(&xs[ti & 1][lane * XSTRIDE]);

        #pragma unroll
        for (int chunk = 0; chunk < Ddim / 32; ++chunk) {   // 8 chunks of 32 floats
            float4 xr[8];
            #pragma unroll
            for (int j = 0; j < 8; ++j) xr[j] = xrow[chunk * 8 + j];

            #pragma unroll
            for (int i = 0; i < 6; ++i) {
                const int b = wid + 8 * i;                  // covers 0..47 across 8 waves
                const float4* __restrict__ orow =
                    (const float4*)(&objs[b * Ddim]);       // same addr across wave -> broadcast
                #pragma unroll
                for (int j = 0; j < 8; ++j) {
                    const float4 o = orow[chunk * 8 + j];
                    acc[i].x += fabsf(o.x - xr[j].x);
                    acc[i].y += fabsf(o.y - xr[j].y);
                    acc[i].z += fabsf(o.z - xr[j].z);
                    acc[i].w += fabsf(o.w - xr[j].w);
                }
            }
        }

        if (n0 + lane < N) {
            #pragma unroll
            for (int i = 0; i < 6; ++i) {
                const int b = wid + 8 * i;
                const float s = (acc[i].x + acc[i].y) + (acc[i].z + acc[i].w);
                score[(size_t)b * N + (n0 + lane)] = GAMMA_F - s;
            }
        }
        // no trailing barrier needed: next iteration's wait+barrier provides the fence
    }
}

__global__ __launch_bounds__(256) void transe_gather_kernel(
    const float* __restrict__ score, const int* __restrict__ idx,
    float* __restrict__ out, int N, int K, int total)
{
    const int i = blockIdx.x * blockDim.x + threadIdx.x;
    if (i < total) {
        const int b = i / K;
        const int e = idx[i];
        out[i] = score[(size_t)b * N + e];
    }
}

extern "C" void kernel_launch(void* const* d_in, const int* in_sizes, int n_in,
                              void* d_out, int out_size, void* d_ws, size_t ws_size,
                              hipStream_t stream)
{
    (void)n_in; (void)d_ws; (void)ws_size; (void)out_size;

    const float* sub = (const float*)d_in[0];   // [48, 256]
    const float* rel = (const float*)d_in[1];   // [48, 256]
    const float* x   = (const float*)d_in[2];   // [N, 256]
    /* d_in[3] = label, unused */
    const int*   pne = (const int*)d_in[4];     // [48, K]

    const int N  = in_sizes[2] / Ddim;          // 32000
    const int BK = in_sizes[4];                 // 48*K
    const int K  = BK / Bdim;                   // 256

    float* out_g = (float*)d_out;               // gathered [B, K] first
    float* out_s = (float*)d_out + BK;          // then original_score [B, N]

    const int tile_span = NT * TPB_TILES;
    const dim3 grid((N + tile_span - 1) / tile_span);
    transe_score_kernel<<<grid, 256, 0, stream>>>(sub, rel, x, out_s, N);
    transe_gather_kernel<<<(BK + 255) / 256, 256, 0, stream>>>(out_s, pne, out_g, N, K, BK);
}